// FC_57002805953244
// MI455X (gfx1250) — compile-verified
//
#include <hip/hip_runtime.h>
#include <hip/hip_bf16.h>

typedef __attribute__((ext_vector_type(2))) float v2f;
typedef __attribute__((ext_vector_type(4))) float v4f;
typedef __attribute__((ext_vector_type(8))) float v8f;

#define LEAKY_SLOPE 0.01f
#define BN_EPS 1e-5f
#define NCHUNK 8          // S-chunks per batch row for the pooling pass
#define POOL_T 192        // threads per pool block: 192 * float4 = 768 = H

// ---------------------------------------------------------------------------
// K0: per-row valid length  L[b] = sum(mask[b,:])
// ---------------------------------------------------------------------------
__global__ void length_kernel(const int* __restrict__ mask,
                              int* __restrict__ Lbuf, int S) {
  const int b = blockIdx.x;
  const int tid = threadIdx.x;            // 256
  __shared__ int red[256];
  int s = 0;
  for (int i = tid; i < S; i += 256) s += mask[b * S + i];
  red[tid] = s;
  __syncthreads();
  for (int off = 128; off > 0; off >>= 1) {
    if (tid < off) red[tid] += red[tid + off];
    __syncthreads();
  }
  if (tid == 0) Lbuf[b] = red[0];
}

// ---------------------------------------------------------------------------
// K1: partial pooled sums.  Grid (NCHUNK, B), 192 threads; each lane owns one
// float4 column group -> one global_load_b128 per token, non-temporal (x is
// read once and is 2x the 192MB L2; don't pollute it).
// ---------------------------------------------------------------------------
__global__ void pool_partial_kernel(const float* __restrict__ x,
                                    const int* __restrict__ Lbuf,
                                    float* __restrict__ partial,
                                    int S, int H, int CH) {
  const int c = blockIdx.x;               // chunk
  const int b = blockIdx.y;               // batch row
  const int f = threadIdx.x;              // float4 column index 0..191
  const int L = Lbuf[b];

  int s0 = c * CH;            if (s0 < 1) s0 = 1;
  int s1 = (c + 1) * CH;      const int e = L - 1; if (s1 > e) s1 = e;

  v4f acc = {0.f, 0.f, 0.f, 0.f};
  const v4f* xb = (const v4f*)(x + (size_t)b * S * H);
  for (int s = s0; s < s1; ++s) {
    v4f v = __builtin_nontemporal_load(xb + (size_t)s * (H / 4) + f);
    acc += v;
  }
  v4f* out = (v4f*)(partial + ((size_t)b * NCHUNK + c) * H);
  out[f] = acc;               // zeros when range empty -> reduce pass is exact
}

// ---------------------------------------------------------------------------
// K2: deterministic chunk reduction + mean scale.  pooled[b,:] = sum/(L-2)
// ---------------------------------------------------------------------------
__global__ void pool_reduce_kernel(const float* __restrict__ partial,
                                   const int* __restrict__ Lbuf,
                                   float* __restrict__ pooled, int H) {
  const int b = blockIdx.x;
  const int f = threadIdx.x;              // 0..191
  v4f acc = {0.f, 0.f, 0.f, 0.f};
  for (int c = 0; c < NCHUNK; ++c) {
    const v4f* p = (const v4f*)(partial + ((size_t)b * NCHUNK + c) * H);
    acc += p[f];
  }
  const float inv = 1.0f / (float)(Lbuf[b] - 2);
  acc *= inv;
  ((v4f*)(pooled + (size_t)b * H))[f] = acc;
}

// ---------------------------------------------------------------------------
// K3: per-column batch stats over M=256 rows (row stride ld) ->
//     mean[c], rstd[c] = rsqrt(var+eps)
// ---------------------------------------------------------------------------
__global__ void colstats_kernel(const float* __restrict__ Y, int ld,
                                float* __restrict__ mean,
                                float* __restrict__ rstd) {
  const int c = blockIdx.x;
  const int r = threadIdx.x;              // 256 == batch size
  const float v = Y[r * ld + c];
  __shared__ float s1[256];
  __shared__ float s2[256];
  s1[r] = v;
  s2[r] = v * v;
  __syncthreads();
  for (int off = 128; off > 0; off >>= 1) {
    if (r < off) { s1[r] += s1[r + off]; s2[r] += s2[r + off]; }
    __syncthreads();
  }
  if (r == 0) {
    const float m = s1[0] * (1.0f / 256.0f);
    const float var = s2[0] * (1.0f / 256.0f) - m * m;   // biased variance
    mean[c] = m;
    rstd[c] = rsqrtf(var + BN_EPS);
  }
}

// ---------------------------------------------------------------------------
// K4: Z = [leaky?]( (Y - mean) * rstd * gamma + beta )  (float4 elementwise)
// Hoists BN out of the GEMM so A is normalized once, not once per N-tile.
// ---------------------------------------------------------------------------
__global__ void bn_apply_kernel(const float* __restrict__ Y,
                                const float* __restrict__ mean,
                                const float* __restrict__ rstd,
                                const float* __restrict__ gamma,
                                const float* __restrict__ beta,
                                int applyLeaky,
                                float* __restrict__ Z,
                                int total4, int K4) {
  const int i = blockIdx.x * blockDim.x + threadIdx.x;
  if (i >= total4) return;
  const int c4 = i % K4;
  v4f y  = ((const v4f*)Y)[i];
  v4f m  = ((const v4f*)mean)[c4];
  v4f r  = ((const v4f*)rstd)[c4];
  v4f gm = ((const v4f*)gamma)[c4];
  v4f bt = ((const v4f*)beta)[c4];
  v4f z = (y - m) * r * gm + bt;
  if (applyLeaky) {
#pragma unroll
    for (int j = 0; j < 4; ++j) z[j] = z[j] > 0.0f ? z[j] : z[j] * LEAKY_SLOPE;
  }
  ((v4f*)Z)[i] = z;
}

// ---------------------------------------------------------------------------
// K5: zero-pad W2 [K x 10] -> W2p [K x 16], b2 -> b2p[16], so the GEMM
// template never needs per-lane bounds predication in its hot loop.
// ---------------------------------------------------------------------------
__global__ void pad_w_kernel(const float* __restrict__ W, const float* __restrict__ b,
                             float* __restrict__ Wp, float* __restrict__ bp,
                             int K, int N, int Np) {
  const int i = blockIdx.x * blockDim.x + threadIdx.x;
  if (i < K * Np) {
    const int n = i % Np;
    Wp[i] = (n < N) ? W[(i / Np) * N + n] : 0.0f;
  }
  if (i < Np) bp[i] = (i < N) ? b[i] : 0.0f;
}

// ---------------------------------------------------------------------------
// K6: Y = Z @ W + bias via V_WMMA_F32_16X16X4_F32, N/K compile-time so all
// strides strength-reduce to immediate offsets; no predication anywhere.
// One wave per 16x16 tile; K stepped by 8 with two independent accumulators
// (no D->C serial chain between consecutive v_wmma).
// fp32 WMMA layout (ISA 7.12.2): lane = 16g+l16; A-reg r = A[m=l16][k0+2g+r];
// B-reg r = W[k0+2g+r][n=l16]; D-reg r = Y[m=8g+r][n=l16].
// ---------------------------------------------------------------------------
template <int M, int N, int K>
__global__ void gemm_wmma_kernel(const float* __restrict__ Z,
                                 const float* __restrict__ W,
                                 const float* __restrict__ bias,
                                 float* __restrict__ Y) {
  constexpr int ntilesN = N / 16;
  constexpr int totalTiles = (M / 16) * ntilesN;
  const int wave = (int)((blockIdx.x * blockDim.x + threadIdx.x) >> 5);
  if (wave >= totalTiles) return;         // uniform per wave (EXEC stays full)

  const int lane = (int)(threadIdx.x & 31);
  const int g   = lane >> 4;
  const int l16 = lane & 15;
  const int mt = wave / ntilesN;
  const int nt = wave % ntilesN;
  const int col = nt * 16 + l16;

  const float* zr = Z + (size_t)(mt * 16 + l16) * K + 2 * g;  // A row base
  const float* wc = W + (size_t)2 * g * N + col;              // B column base

  v8f acc0 = {};
  v8f acc1 = {};
  for (int k0 = 0; k0 < K; k0 += 8) {     // K in {768,512,256}
    v2f a0 = *(const v2f*)(zr + k0);      // global_load_b64
    v2f a1 = *(const v2f*)(zr + k0 + 4);
    v2f b0, b1;
    b0.x = wc[(k0    ) * N];
    b0.y = wc[(k0 + 1) * N];
    b1.x = wc[(k0 + 4) * N];
    b1.y = wc[(k0 + 5) * N];
    acc0 = __builtin_amdgcn_wmma_f32_16x16x4_f32(false, a0, false, b0,
                                                 (short)0, acc0, false, false);
    acc1 = __builtin_amdgcn_wmma_f32_16x16x4_f32(false, a1, false, b1,
                                                 (short)0, acc1, false, false);
  }

  const float bb = bias[col];
#pragma unroll
  for (int r = 0; r < 8; ++r) {
    const int orow = mt * 16 + r + 8 * g;
    Y[(size_t)orow * N + col] = acc0[r] + acc1[r] + bb;
  }
}

// ---------------------------------------------------------------------------
// K7: final  out[b,c] = leaky(BN(Y[b*ld + c]))  (reads padded y2, ld=16)
// ---------------------------------------------------------------------------
__global__ void bn_act_out_kernel(const float* __restrict__ Y,
                                  const float* __restrict__ mean,
                                  const float* __restrict__ rstd,
                                  const float* __restrict__ gamma,
                                  const float* __restrict__ beta,
                                  float* __restrict__ out,
                                  int total, int N, int ld) {
  const int i = blockIdx.x * blockDim.x + threadIdx.x;
  if (i >= total) return;
  const int b = i / N;
  const int c = i % N;
  float v = (Y[b * ld + c] - mean[c]) * rstd[c] * gamma[c] + beta[c];
  out[i] = v > 0.0f ? v : v * LEAKY_SLOPE;
}

// ---------------------------------------------------------------------------
extern "C" void kernel_launch(void* const* d_in, const int* in_sizes, int n_in,
                              void* d_out, int out_size, void* d_ws, size_t ws_size,
                              hipStream_t stream) {
  constexpr int B = 256, S = 512, H = 768, D1 = 512, D2 = 256, C = 10, CP = 16;
  constexpr int CH = S / NCHUNK;          // 64 tokens per chunk

  const float* x    = (const float*)d_in[0];
  const int*   mask = (const int*)  d_in[1];
  const float* W0   = (const float*)d_in[2];
  const float* b0   = (const float*)d_in[3];
  const float* W1   = (const float*)d_in[4];
  const float* b1   = (const float*)d_in[5];
  const float* W2   = (const float*)d_in[6];
  const float* b2   = (const float*)d_in[7];
  const float* g_e  = (const float*)d_in[8];
  const float* be_e = (const float*)d_in[9];
  const float* g0   = (const float*)d_in[10];
  const float* bt0  = (const float*)d_in[11];
  const float* g1   = (const float*)d_in[12];
  const float* bt1  = (const float*)d_in[13];
  const float* g2   = (const float*)d_in[14];
  const float* bt2  = (const float*)d_in[15];
  float* out = (float*)d_out;

  // workspace layout (floats)
  float* ws      = (float*)d_ws;
  int*   Lbuf    = (int*)ws;                          // 256 ints
  float* partial = ws + 256;                          // B*NCHUNK*H = 1,572,864
  float* pooled  = partial + (size_t)B * NCHUNK * H;  // 196,608
  float* z0      = pooled + B * H;                    // 196,608
  float* y0      = z0 + B * H;                        // 131,072
  float* z1      = y0 + B * D1;                       // 131,072
  float* y1      = z1 + B * D1;                       //  65,536
  float* z2      = y1 + B * D2;                       //  65,536
  float* y2      = z2 + B * D2;                       // B*CP = 4,096
  float* W2p     = y2 + B * CP;                       // D2*CP = 4,096
  float* b2p     = W2p + D2 * CP;                     // 16
  float* mean_p  = b2p + 16;
  float* rstd_p  = mean_p + H;
  float* mean0   = rstd_p + H;
  float* rstd0   = mean0 + D1;
  float* mean1   = rstd0 + D1;
  float* rstd1   = mean1 + D2;
  float* mean2   = rstd1 + D2;
  float* rstd2   = mean2 + 16;
  (void)ws_size; (void)n_in; (void)in_sizes; (void)out_size; (void)rstd2;

  // --- pooling (the only HBM-scale work: ~400 MB of x) ---
  length_kernel<<<B, 256, 0, stream>>>(mask, Lbuf, S);
  pool_partial_kernel<<<dim3(NCHUNK, B), POOL_T, 0, stream>>>(
      x, Lbuf, partial, S, H, CH);
  pool_reduce_kernel<<<B, POOL_T, 0, stream>>>(partial, Lbuf, pooled, H);

  // --- pad last-layer weights once (16 KB) so no GEMM needs predication ---
  pad_w_kernel<<<(D2 * CP + 255) / 256, 256, 0, stream>>>(
      W2, b2, W2p, b2p, D2, C, CP);

  // --- layer 0: h = BN(pooled);  y0 = h @ W0 + b0 ---
  colstats_kernel<<<H, 256, 0, stream>>>(pooled, H, mean_p, rstd_p);
  bn_apply_kernel<<<(B * H / 4 + 255) / 256, 256, 0, stream>>>(
      pooled, mean_p, rstd_p, g_e, be_e, /*leaky=*/0, z0, B * H / 4, H / 4);
  gemm_wmma_kernel<B, D1, H><<<(B / 16) * (D1 / 16) / 8, 256, 0, stream>>>(
      z0, W0, b0, y0);

  // --- layer 1: y1 = leaky(BN(y0)) @ W1 + b1 ---
  colstats_kernel<<<D1, 256, 0, stream>>>(y0, D1, mean0, rstd0);
  bn_apply_kernel<<<(B * D1 / 4 + 255) / 256, 256, 0, stream>>>(
      y0, mean0, rstd0, g0, bt0, /*leaky=*/1, z1, B * D1 / 4, D1 / 4);
  gemm_wmma_kernel<B, D2, D1><<<(B / 16) * (D2 / 16) / 8, 256, 0, stream>>>(
      z1, W1, b1, y1);

  // --- layer 2 (padded N=16): y2 = leaky(BN(y1)) @ W2p + b2p ---
  colstats_kernel<<<D2, 256, 0, stream>>>(y1, D2, mean1, rstd1);
  bn_apply_kernel<<<(B * D2 / 4 + 255) / 256, 256, 0, stream>>>(
      y1, mean1, rstd1, g1, bt1, /*leaky=*/1, z2, B * D2 / 4, D2 / 4);
  gemm_wmma_kernel<B, CP, D2><<<(B / 16) * (CP / 16) / 8, 256, 0, stream>>>(
      z2, W2p, b2p, y2);

  // --- output: out = leaky(BN(y2[:, :10])) ---
  colstats_kernel<<<C, 256, 0, stream>>>(y2, CP, mean2, rstd2);
  bn_act_out_kernel<<<(B * C + 255) / 256, 256, 0, stream>>>(
      y2, mean2, rstd2, g2, bt2, out, B * C, C, CP);
}